// WeightedEdgeSoftmax_14336600834853
// MI455X (gfx1250) — compile-verified
//
#include <hip/hip_runtime.h>
#include <hip/hip_bf16.h>

// Weighted edge softmax for MI455X (gfx1250).
//   pass1: per-(node,head) segmented max via single-instruction u32 atomics
//          (order-preserving float->uint encoding; scratch is L2-resident).
//   pass2: e = scale * exp(logit - max[dst]); contiguous logits/scale streams
//          are double-buffered into LDS with gfx1250 async-to-LDS loads
//          (ASYNCcnt pipeline), while the VGPR/LOADcnt path handles the
//          irregular dst/max gathers. Output stored non-temporal.
// No matmul exists in this op => no WMMA by design; this targets the CDNA5
// memory paths: global_atomic_max_u32, global_prefetch_b8,
// global_load_async_to_lds_b128 + s_wait_asynccnt, NT temporal hints.

#define HEADS 8
#define TPB   256   // 8 wave32 waves per block
#define SLOT  12    // floats per LDS slot: 48B = 16B-aligned + bank-spread

typedef float    v4f __attribute__((ext_vector_type(4)));
typedef unsigned v4u __attribute__((ext_vector_type(4)));

// Order-preserving float -> uint encoding: unsigned max == float max.
__device__ __forceinline__ unsigned f2ord(float f) {
    unsigned u = __float_as_uint(f);
    return (u & 0x80000000u) ? ~u : (u | 0x80000000u);
}
__device__ __forceinline__ float ord2f(unsigned u) {
    return (u & 0x80000000u) ? __uint_as_float(u & 0x7fffffffu)
                             : __uint_as_float(~u);
}

// Low 32 bits of a generic pointer to __shared__ == LDS byte address.
__device__ __forceinline__ unsigned lds_lo32(const void* p) {
    return (unsigned)(uintptr_t)p;
}

// Async-copy 32 contiguous bytes (8 floats) from global to this lane's LDS
// slot: two b128 transfers. The offset: immediate applies to BOTH the LDS
// destination and the global address (ISA 10.x async pseudocode), which is
// exactly the hi-half placement we need.
__device__ __forceinline__ void async_copy32(unsigned lds_off, unsigned byte_off,
                                             const float* base) {
    asm volatile("global_load_async_to_lds_b128 %0, %1, %2"
                 :: "v"(lds_off), "v"(byte_off), "s"(base) : "memory");
    asm volatile("global_load_async_to_lds_b128 %0, %1, %2 offset:16"
                 :: "v"(lds_off), "v"(byte_off), "s"(base) : "memory");
}

// ---------------------------------------------------------------- kernel 1
// Initialize the per-(node,head) running max to orderable -inf (== 0u).
__global__ void ws_init_kernel(unsigned* __restrict__ ws,
                               const int* __restrict__ num_nodes_p) {
    const int total = num_nodes_p[0] * HEADS;
    for (int i = blockIdx.x * blockDim.x + threadIdx.x; i < total;
         i += gridDim.x * blockDim.x) {
        ws[i] = 0u;
    }
}

// ---------------------------------------------------------------- kernel 2
// Segmented max: one thread per edge, 2 x b128 loads of the 8 head logits,
// 8 non-returning global_atomic_max_u32 into the 3.2MB L2-resident scratch.
// Loads stay RT so logits remain in L2 for the second pass.
__global__ void seg_max_kernel(const float* __restrict__ logits,
                               const int*  __restrict__ dst,
                               unsigned*   __restrict__ ws,
                               int num_edges) {
    const int e = blockIdx.x * blockDim.x + threadIdx.x;
    if (e >= num_edges) return;

    const int pe = e + 8 * TPB * 32;   // deep streaming prefetch into L2
    if (pe < num_edges) {
        __builtin_prefetch(logits + (size_t)pe * HEADS, 0, 3);
    }

    const int d = dst[e];
    const v4f* lp = (const v4f*)(logits + (size_t)e * HEADS);
    const v4f la = lp[0];
    const v4f lb = lp[1];

    unsigned* m = ws + (size_t)d * HEADS;
    atomicMax(m + 0, f2ord(la.x));
    atomicMax(m + 1, f2ord(la.y));
    atomicMax(m + 2, f2ord(la.z));
    atomicMax(m + 3, f2ord(la.w));
    atomicMax(m + 4, f2ord(lb.x));
    atomicMax(m + 5, f2ord(lb.y));
    atomicMax(m + 6, f2ord(lb.z));
    atomicMax(m + 7, f2ord(lb.w));
}

// ---------------------------------------------------------------- kernel 3
// Emit: e = scale * exp(logit - max[dst]).
// Persistent grid-stride; logits/scale tiles double-buffered into LDS with
// async-to-LDS b128 loads; irregular dst/max gathers stay on the VGPR path;
// NT b128 stores for the single-use output.
__global__ void __launch_bounds__(TPB)
emit_kernel(const float*    __restrict__ logits,
            const float*    __restrict__ scale,
            const int*      __restrict__ dst,
            const unsigned* __restrict__ ws,
            float*          __restrict__ out,
            int num_edges) {
    __shared__ __align__(16) float lbuf[2][TPB][SLOT];
    __shared__ __align__(16) float sbuf[2][TPB][SLOT];

    const int tid = threadIdx.x;
    const int S   = gridDim.x * TPB;          // edge stride per iteration
    int base      = blockIdx.x * TPB;         // block-uniform tile base
    int e         = base + tid;

    // Issue the 4 async b128s (32B logits + 32B scale) for one edge.
    auto issue = [&](int edge, int b) {
        const unsigned off = (unsigned)edge * (HEADS * 4u);
        async_copy32(lds_lo32(&lbuf[b][tid][0]), off, logits);
        async_copy32(lds_lo32(&sbuf[b][tid][0]), off, scale);
    };

    if (base < num_edges) {
        const int ecl = (e < num_edges) ? e : (num_edges - 1);  // clamp tail
        issue(ecl, 0);
    }

    int buf = 0;
    for (; base < num_edges; base += S, e += S, buf ^= 1) {
        const bool have_next = (base + S) < num_edges;   // wave-uniform
        if (have_next) {
            const int en  = e + S;
            const int ecl = (en < num_edges) ? en : (num_edges - 1);
            issue(ecl, buf ^ 1);
            // 8 outstanding; in-order completion => first 4 (current tile)
            // are done once ASYNCcnt <= 4.
            asm volatile("s_wait_asynccnt 0x4" ::: "memory");
        } else {
            asm volatile("s_wait_asynccnt 0x0" ::: "memory");
        }

        if (e < num_edges) {
            const v4f la = *(const v4f*)&lbuf[buf][tid][0];
            const v4f lb = *(const v4f*)&lbuf[buf][tid][4];
            const v4f sa = *(const v4f*)&sbuf[buf][tid][0];
            const v4f sb = *(const v4f*)&sbuf[buf][tid][4];

            const int d = dst[e];
            const v4u* mp = (const v4u*)(ws + (size_t)d * HEADS);
            const v4u ma = mp[0];
            const v4u mb = mp[1];

            v4f r0, r1;
            r0.x = sa.x * __expf(la.x - ord2f(ma.x));
            r0.y = sa.y * __expf(la.y - ord2f(ma.y));
            r0.z = sa.z * __expf(la.z - ord2f(ma.z));
            r0.w = sa.w * __expf(la.w - ord2f(ma.w));
            r1.x = sb.x * __expf(lb.x - ord2f(mb.x));
            r1.y = sb.y * __expf(lb.y - ord2f(mb.y));
            r1.z = sb.z * __expf(lb.z - ord2f(mb.z));
            r1.w = sb.w * __expf(lb.w - ord2f(mb.w));

            v4f* op = (v4f*)(out + (size_t)e * HEADS);
            __builtin_nontemporal_store(r0, op);
            __builtin_nontemporal_store(r1, op + 1);
        }
    }
}

extern "C" void kernel_launch(void* const* d_in, const int* in_sizes, int n_in,
                              void* d_out, int out_size, void* d_ws, size_t ws_size,
                              hipStream_t stream) {
    const float* logits    = (const float*)d_in[0];   // [E, H, 1] f32
    const float* scale     = (const float*)d_in[1];   // [E, H]    f32
    const int*   dst       = (const int*)  d_in[2];   // [E]       i32
    const int*   num_nodes = (const int*)  d_in[3];   // scalar    i32 (device)
    float*       out       = (float*)d_out;           // [E, H, 1] f32
    unsigned*    ws        = (unsigned*)d_ws;         // N*H u32 running max

    const int num_edges = in_sizes[2];
    const int blocks    = (num_edges + TPB - 1) / TPB;

    // Persistent emit grid: enough blocks for full occupancy, few enough that
    // each wave runs a multi-tile async pipeline.
    int eblocks = blocks < 3072 ? blocks : 3072;

    ws_init_kernel<<<2048, TPB, 0, stream>>>(ws, num_nodes);
    seg_max_kernel<<<blocks, TPB, 0, stream>>>(logits, dst, ws, num_edges);
    emit_kernel<<<eblocks, TPB, 0, stream>>>(logits, scale, dst, ws, out, num_edges);
}